// PairEmb_78185584656591
// MI455X (gfx1250) — compile-verified
//
#include <hip/hip_runtime.h>

// Reference problem constants: B=32, T=2048, D=256, P=100000, MAX_SPAN=16.
// token_embs (64 MB) is fully L2-resident on MI455X (192 MB L2); output is
// 205 MB -> ~9us HBM floor. The bound is L2 read volume, so we load exactly
// ceil(span/4)*4 window rows per pair-side (wave-uniform step count -> scalar
// branch, no exec divergence), and the span-sum runs on the matrix unit as
// D = A(mask 16x4) x B(window 4x16), chained V_WMMA_F32_16X16X4_F32.

typedef __attribute__((ext_vector_type(2))) float v2f;
typedef __attribute__((ext_vector_type(8))) float v8f;

#define TDIM 2048
#define DDIM 256

// One pair-side, NSTEPS K=4 chain steps per 16-column tile (NSTEPS uniform
// across the wave). Rows beyond span inside the last step are zeroed by the
// A-mask; rows beyond 4*NSTEPS are neither loaded nor accumulated.
template <int NSTEPS>
__device__ __forceinline__ void span_mean_tiles(
    const float* __restrict__ lanebase,  // embs row window base + this lane's col
    float*       __restrict__ outp,      // output row for this pair-side
    const v2f   (&afrag)[4],             // mask fragments (only [0..NSTEPS) used)
    float inv, int lane, int col)
{
    #pragma unroll 8
    for (int t = 0; t < 16; ++t) {
        v8f acc = {};
        #pragma unroll
        for (int c = 0; c < NSTEPS; ++c) {
            // Per ISA 7.12.2 striping: this lane's float2 covers rows
            // (4c+rh, 4c+rh+1), rh = (lane<16 ? 0 : 2), already folded into
            // lanebase's rh offset below via afrag/bfrag symmetry.
            v2f bfrag;
            bfrag.x = lanebase[(4 * c    ) * DDIM + t * 16];
            bfrag.y = lanebase[(4 * c + 1) * DDIM + t * 16];
            acc = __builtin_amdgcn_wmma_f32_16x16x4_f32(
                false, afrag[c], false, bfrag, (short)0, acc, false, false);
        }
        // D rows replicated; row M=0 = acc[0], lanes 0-15 (N = lane).
        if (lane < 16)
            outp[t * 16 + col] = acc[0] * inv;
    }
}

__global__ __launch_bounds__(256) void PairEmb_span_mean_wmma(
    const float* __restrict__ embs,      // [B, T, D] f32
    const int*   __restrict__ p1_start,  // [P]
    const int*   __restrict__ p1_span,   // [P]
    const int*   __restrict__ p2_start,  // [P]
    const int*   __restrict__ p2_span,   // [P]
    const int*   __restrict__ pair_batch,// [P]
    float*       __restrict__ out,       // [P, 2*D]
    int P)
{
    const int lane = threadIdx.x & 31;
    const int wave = threadIdx.x >> 5;          // 8 waves per block
    const int ps   = blockIdx.x * 8 + wave;     // pair-side id in [0, 2P)
    if (ps >= 2 * P) return;                    // wave-uniform: EXEC all-1s for WMMA

    const int p    = ps >> 1;
    const int side = ps & 1;

    const int b     = pair_batch[p];
    const int start = side ? p2_start[p] : p1_start[p];
    const int span  = side ? p2_span[p]  : p1_span[p];
    const float inv = 1.0f / (float)span;

    // Lane-half row offset per the 32-bit A/B fragment layout.
    const int rh  = (lane < 16) ? 0 : 2;
    const int col = lane & 15;

    // Mask matrix A: ones for k < span, replicated across all 16 M rows.
    v2f afrag[4];
    #pragma unroll
    for (int c = 0; c < 4; ++c) {
        const int k0 = 4 * c + rh;
        afrag[c].x = (k0     < span) ? 1.0f : 0.0f;
        afrag[c].y = (k0 + 1 < span) ? 1.0f : 0.0f;
    }

    // Per-lane base: rows [start, start+16) always in-bounds (start < T-16).
    // rh folds the lane-half's +2-row shift into the base so every load below
    // is a compile-time immediate offset (max < 16 KB << 2^23).
    const float* lanebase =
        embs + ((size_t)b * TDIM + start + rh) * DDIM + col;

    float* outp = out + (size_t)p * (2 * DDIM) + (size_t)side * DDIM;

    // Wave-uniform step count: skip whole K=4 steps (loads AND wmmas) whose
    // mask would be all-zero. Scalar branch -> no exec divergence.
    const int nsteps = (span + 3) >> 2;   // span in [1,16] -> nsteps in [1,4]
    switch (nsteps) {
        case 1:  span_mean_tiles<1>(lanebase, outp, afrag, inv, lane, col); break;
        case 2:  span_mean_tiles<2>(lanebase, outp, afrag, inv, lane, col); break;
        case 3:  span_mean_tiles<3>(lanebase, outp, afrag, inv, lane, col); break;
        default: span_mean_tiles<4>(lanebase, outp, afrag, inv, lane, col); break;
    }
}

extern "C" void kernel_launch(void* const* d_in, const int* in_sizes, int n_in,
                              void* d_out, int out_size, void* d_ws, size_t ws_size,
                              hipStream_t stream) {
    const float* embs     = (const float*)d_in[0];
    const int*   p1_start = (const int*)  d_in[1];
    const int*   p1_span  = (const int*)  d_in[2];
    const int*   p2_start = (const int*)  d_in[3];
    const int*   p2_span  = (const int*)  d_in[4];
    const int*   pbatch   = (const int*)  d_in[5];
    float*       out      = (float*)      d_out;

    const int P = in_sizes[1];                 // p1_start has P elements
    const int pair_sides = 2 * P;              // one wave32 per pair-side
    const int blocks = (pair_sides + 7) / 8;   // 8 waves per 256-thread block

    PairEmb_span_mean_wmma<<<blocks, 256, 0, stream>>>(
        embs, p1_start, p1_span, p2_start, p2_span, pbatch, out, P);
}